// DNNmodel_74045236183344
// MI455X (gfx1250) — compile-verified
//
#include <hip/hip_runtime.h>
#include <hip/hip_bf16.h>

// 1024 independent scalar LSTM chains (I=H=O=1), T=16384 sequential steps.
// Latency-bound on the per-step dependency chain; only 32 wave32 waves exist.
// CDNA5 paths used:
//   - V_TANH_F32 hardware transcendental (sigmoid folded into tanh, 0.5 scale
//     pre-baked into scalar weights) -> shortest possible per-step chain.
//   - Tensor Data Mover: one tensor_load_to_lds per 32x128 tile of x
//     (32 batch rows, stride T) double-buffered in LDS, s_wait_tensorcnt
//     synchronization. TDM pad feature (4 dwords per 512B row) makes the
//     per-lane ds_load_b128 row reads LDS-bank-conflict-free.

#ifndef __has_builtin
#define __has_builtin(x) 0
#endif

__device__ __forceinline__ float fast_tanh(float x) {
#if __has_builtin(__builtin_amdgcn_tanhf)
  return __builtin_amdgcn_tanhf(x);
#else
  float r;
  asm("v_tanh_f32 %0, %1" : "=v"(r) : "v"(x));
  return r;
#endif
}

#define LSTM_B 1024
#define LSTM_T 16384

#if __has_builtin(__builtin_amdgcn_tensor_load_to_lds) && \
    __has_builtin(__builtin_amdgcn_s_wait_tensorcnt)
#define USE_TDM 1
#else
#define USE_TDM 0
#endif

#if USE_TDM
typedef __attribute__((ext_vector_type(4))) unsigned int v4u;
typedef __attribute__((ext_vector_type(8))) int v8i;
typedef __attribute__((ext_vector_type(4))) int v4i;

#define CHUNK 128                  // timesteps per TDM tile
#define NCH   (LSTM_T / CHUNK)     // 128 chunks
#define ROWDW (CHUNK + 4)          // 128 data dwords + 4 pad dwords (TDM pad)
#endif

// One recurrence step. 0.5 factors for i/f/o gates are pre-folded into the
// *_h weights/biases: sigmoid(z) = 0.5*tanh(z/2)+0.5.  Uses locals:
//   h, c (state), wih_*, whh_*, bs_* (folded params), W00 (out weight),
//   yb (out bias).
#define LSTM_STEP(XS, YS)                                                  \
  do {                                                                     \
    float gi = fmaf(h, whh_i, fmaf((XS), wih_i, bs_i));                    \
    float gf = fmaf(h, whh_f, fmaf((XS), wih_f, bs_f));                    \
    float gg = fmaf(h, whh_g, fmaf((XS), wih_g, bs_g));                    \
    float go = fmaf(h, whh_o, fmaf((XS), wih_o, bs_o));                    \
    float ti = fast_tanh(gi);                                              \
    float tf = fast_tanh(gf);                                              \
    float tg = fast_tanh(gg);                                              \
    float to = fast_tanh(go);                                              \
    float ig = fmaf(ti, 0.5f, 0.5f) * tg;    /* i * g         */           \
    c = fmaf(fmaf(tf, 0.5f, 0.5f), c, ig);   /* c = f*c + i*g */           \
    h = fmaf(to, 0.5f, 0.5f) * fast_tanh(c); /* h = o*tanh(c) */           \
    (YS) = fmaf(h, W00, yb);                 /* off critical path */       \
  } while (0)

__global__ __launch_bounds__(32) void lstm_scalar_chain(
    const float* __restrict__ x,     // [B, T]
    const float* __restrict__ w_ih,  // [4]  (i, f, g, o)
    const float* __restrict__ w_hh,  // [4]
    const float* __restrict__ b_ih,  // [4]
    const float* __restrict__ b_hh,  // [4]
    const float* __restrict__ Wp,    // [1]
    const float* __restrict__ bp,    // [1]
    float* __restrict__ y)           // [B, T]
{
  const int lane = threadIdx.x;
  const int bb0  = blockIdx.x * 32;   // batch base of this wave
  const int b    = bb0 + lane;        // this lane's batch/chain

  const float wih_i = 0.5f * w_ih[0], whh_i = 0.5f * w_hh[0];
  const float wih_f = 0.5f * w_ih[1], whh_f = 0.5f * w_hh[1];
  const float wih_g =        w_ih[2], whh_g =        w_hh[2];
  const float wih_o = 0.5f * w_ih[3], whh_o = 0.5f * w_hh[3];
  const float bs_i  = 0.5f * (b_ih[0] + b_hh[0]);
  const float bs_f  = 0.5f * (b_ih[1] + b_hh[1]);
  const float bs_g  =        (b_ih[2] + b_hh[2]);
  const float bs_o  = 0.5f * (b_ih[3] + b_hh[3]);
  const float W00 = Wp[0];
  const float yb  = bp[0];

  float h = 0.0f, c = 0.0f;

#if USE_TDM
  // Double-buffered x tiles: 32 rows x (128 + 4 pad) dwords each.
  __shared__ float xbuf[2][32 * ROWDW];  // 2 * 16896 B = 33792 B LDS

  const unsigned int lds_base  = (unsigned int)(uintptr_t)(&xbuf[0][0]);
  const unsigned int buf_bytes = (unsigned int)sizeof(xbuf[0]);
  const float* gbase = x + (size_t)bb0 * LSTM_T;  // tile row 0 = batch bb0

  // Issue one TDM 2D tile load: rows = 32 batches (stride T floats),
  // cols = CHUNK floats, into LDS buffer `sel` with 4-dword row padding.
  auto tdm_issue = [&](int ch, int sel) {
    unsigned long long ga =
        (unsigned long long)(uintptr_t)(gbase + (size_t)ch * CHUNK);
    v4u g0;
    g0[0] = 1u;                                        // count=1, user D#
    g0[1] = lds_base + (unsigned int)sel * buf_bytes;  // lds_addr
    g0[2] = (unsigned int)ga;                          // global_addr[31:0]
    g0[3] = (unsigned int)(ga >> 32) | 0x80000000u;    // addr[56:32] | type=2
    v8i g1;
    g1[0] = (int)0x07920000;   // data_size=4B, pad_en, interval=512B, pad=4dw
    g1[1] = (int)0x40000000;   // tensor_dim0[15:0]=16384 in bits[31:16]
    g1[2] = (int)0x04000000;   // tensor_dim0 hi=0 | tensor_dim1[15:0]=1024
    g1[3] = (int)0x00800000;   // tensor_dim1 hi=0 | tile_dim0=128
    g1[4] = 32;                // tile_dim1=32 rows | tile_dim2=0
    g1[5] = LSTM_T;            // tensor_dim0_stride lo = 16384
    g1[6] = 0;
    g1[7] = 0;
    v4i z4 = {0, 0, 0, 0};           // groups 2/3: tile_dim3/4 = 0 (unused)
    v8i z8 = {0, 0, 0, 0, 0, 0, 0, 0};
    __builtin_amdgcn_tensor_load_to_lds(g0, g1, z4, z4, z8, 0);
  };

  tdm_issue(0, 0);
  tdm_issue(1, 1);

  for (int n = 0; n < NCH; ++n) {
    if (n == NCH - 1) {
      __builtin_amdgcn_s_wait_tensorcnt(0);   // last tile: drain fully
    } else {
      __builtin_amdgcn_s_wait_tensorcnt(1);   // tile n complete
    }
    asm volatile("" ::: "memory");  // LDS now holds tile n

    const float4* __restrict__ xr =
        reinterpret_cast<const float4*>(&xbuf[n & 1][lane * ROWDW]);
    float4* __restrict__ yr = reinterpret_cast<float4*>(
        y + (size_t)b * LSTM_T + (size_t)n * CHUNK);

    for (int q = 0; q < CHUNK / 4; ++q) {
      float4 xq = xr[q];
      float4 yq;
      LSTM_STEP(xq.x, yq.x);
      LSTM_STEP(xq.y, yq.y);
      LSTM_STEP(xq.z, yq.z);
      LSTM_STEP(xq.w, yq.w);
      yr[q] = yq;
    }

    if (n + 2 < NCH) {
      // All ds_loads of this buffer are consumed; make WAR explicit, then
      // refill this buffer with tile n+2.
      asm volatile("s_wait_dscnt 0x0" ::: "memory");
      tdm_issue(n + 2, n & 1);
    }
  }
#else  // !USE_TDM — fallback: per-lane b128 stream + global_prefetch
  const float4* __restrict__ xv =
      reinterpret_cast<const float4*>(x + (size_t)b * LSTM_T);
  float4* __restrict__ yv =
      reinterpret_cast<float4*>(y + (size_t)b * LSTM_T);
  for (int q = 0; q < LSTM_T / 4; ++q) {
    float4 xq = xv[q];
    if ((q & 7) == 0) __builtin_prefetch((const void*)(xv + q + 8), 0, 0);
    float4 yq;
    LSTM_STEP(xq.x, yq.x);
    LSTM_STEP(xq.y, yq.y);
    LSTM_STEP(xq.z, yq.z);
    LSTM_STEP(xq.w, yq.w);
    yv[q] = yq;
  }
#endif
}

extern "C" void kernel_launch(void* const* d_in, const int* in_sizes, int n_in,
                              void* d_out, int out_size, void* d_ws, size_t ws_size,
                              hipStream_t stream) {
  (void)in_sizes; (void)n_in; (void)out_size; (void)d_ws; (void)ws_size;

  const float* x    = (const float*)d_in[0];
  const float* w_ih = (const float*)d_in[1];
  const float* w_hh = (const float*)d_in[2];
  const float* b_ih = (const float*)d_in[3];
  const float* b_hh = (const float*)d_in[4];
  const float* W    = (const float*)d_in[5];
  const float* bb   = (const float*)d_in[6];
  float* y = (float*)d_out;

  // 32 blocks x 1 wave: each wave alone on a SIMD, one LSTM chain per lane.
  dim3 grid(LSTM_B / 32), block(32);
  lstm_scalar_chain<<<grid, block, 0, stream>>>(x, w_ih, w_hh, b_ih, b_hh, W, bb, y);
}